// VectorQuantizer_36618891165961
// MI455X (gfx1250) — compile-verified
//
#include <hip/hip_runtime.h>
#include <hip/hip_bf16.h>

typedef float v2f __attribute__((ext_vector_type(2)));
typedef float v8f __attribute__((ext_vector_type(8)));

#define DIMV    64
#define NEMB    512
#define WAVES   8
#define CHUNK   128   // codes staged in LDS at a time
#define PAD     68    // LDS row stride (floats) to dodge bank conflicts

// ---------------------------------------------------------------------------
// Kernel 0: per-code squared norms  eSq[j] = sum_d embed[d,j]^2
// ---------------------------------------------------------------------------
__global__ __launch_bounds__(512) void vq_esq_kernel(const float* __restrict__ embed,
                                                     float* __restrict__ eSq) {
    int j = threadIdx.x;
    if (j < NEMB) {
        float s = 0.0f;
        #pragma unroll 8
        for (int d = 0; d < DIMV; ++d) {
            float e = embed[d * NEMB + j];
            s += e * e;
        }
        eSq[j] = s;
    }
}

// ---------------------------------------------------------------------------
// Kernel 1: distances via fp32 WMMA, argmin, quantize gather, histogram,
//           segment-sum (embed_sum), diff partial sums.
// One wave handles a 16-row tile; block = 8 waves = 128 rows.
// ---------------------------------------------------------------------------
__global__ __launch_bounds__(256) void vq_dist_kernel(
    const float* __restrict__ x,
    const float* __restrict__ embed,
    const float* __restrict__ eSq,
    float* __restrict__ hist,
    float* __restrict__ embedSum,
    float* __restrict__ diffAcc,
    float* __restrict__ outQ,
    float* __restrict__ outInd) {

    __shared__ float sEmbT[CHUNK][PAD];   // transposed codebook chunk: [code][dim]
    __shared__ float sEsq[CHUNK];
    __shared__ int   sIdx[WAVES * 16];

    const int tid  = threadIdx.x;
    const int wid  = tid >> 5;     // wave 0..7
    const int lane = tid & 31;
    const int l16  = lane & 15;
    const int hi   = lane >> 4;    // 0: K even pair, 1: K odd pair (A/B frag halves)

    const int rowBase = blockIdx.x * (WAVES * 16) + wid * 16;
    const int myRow   = rowBase + l16;

    // --- A fragments: lane holds 2 consecutive dims per K-step (16x4 fp32 layout) ---
    v2f a[16];
    const float* xr = x + (size_t)myRow * DIMV + 2 * hi;
    #pragma unroll
    for (int k = 0; k < 16; ++k)
        a[k] = *(const v2f*)(xr + 4 * k);

    float bestVal[8];
    int   bestIdx[8];
    #pragma unroll
    for (int v = 0; v < 8; ++v) { bestVal[v] = 3.4e38f; bestIdx[v] = 0; }

    for (int c = 0; c < NEMB / CHUNK; ++c) {
        const int jc = c * CHUNK;
        __syncthreads();   // previous chunk fully consumed before overwrite
        // stage transposed chunk: sEmbT[j'][d] = embed[d*512 + jc + j']
        for (int i = tid; i < CHUNK * DIMV; i += 256) {
            int jj = i & (CHUNK - 1);
            int d  = i >> 7;
            sEmbT[jj][d] = embed[d * NEMB + jc + jj];
        }
        if (tid < CHUNK) sEsq[tid] = eSq[jc + tid];
        __syncthreads();

        #pragma unroll 1
        for (int t = 0; t < CHUNK / 16; ++t) {       // 8 N-tiles of 16 codes
            const int jrow = t * 16 + l16;           // lane's code within chunk
            const float* bp = &sEmbT[jrow][2 * hi];  // B fragment base
            v8f acc = {};
            #pragma unroll
            for (int k = 0; k < 16; ++k) {           // K = 64 in steps of 4
                v2f b = *(const v2f*)(bp + 4 * k);
                acc = __builtin_amdgcn_wmma_f32_16x16x4_f32(
                        false, a[k], false, b, (short)0, acc, false, false);
            }
            const float es   = sEsq[jrow];
            const int   jglb = jc + jrow;
            #pragma unroll
            for (int v = 0; v < 8; ++v) {
                float cand = es - 2.0f * acc[v];     // ||f||^2 dropped: row-constant
                if (cand < bestVal[v]) { bestVal[v] = cand; bestIdx[v] = jglb; }
            }
        }
    }

    // --- argmin across the 16 lanes of each half (rows v / v+8), ties -> lower idx ---
    #pragma unroll
    for (int v = 0; v < 8; ++v) {
        float bv = bestVal[v]; int bi = bestIdx[v];
        #pragma unroll
        for (int off = 1; off < 16; off <<= 1) {
            float ov = __shfl_xor(bv, off, 32);
            int   oi = __shfl_xor(bi, off, 32);
            if (ov < bv || (ov == bv && oi < bi)) { bv = ov; bi = oi; }
        }
        bestVal[v] = bv; bestIdx[v] = bi;
    }
    if (l16 == 0) {   // lane 0 -> rows 0..7, lane 16 -> rows 8..15
        #pragma unroll
        for (int v = 0; v < 8; ++v)
            sIdx[wid * 16 + hi * 8 + v] = bestIdx[v];
    }
    __syncthreads();

    const int myIdx = sIdx[wid * 16 + l16];   // winning code of row myRow

    if (hi == 0) {
        outInd[myRow] = (float)myIdx;
        atomicAdd(&hist[myIdx], 1.0f);
    }

    // --- quantize write (straight-through value == code vector), diff, embed_sum ---
    float dsum = 0.0f;
    float* qr = outQ + (size_t)myRow * DIMV + 2 * hi;
    #pragma unroll
    for (int k = 0; k < 16; ++k) {
        int   d0 = 4 * k + 2 * hi;
        float q0 = embed[(d0 + 0) * NEMB + myIdx];
        float q1 = embed[(d0 + 1) * NEMB + myIdx];
        float x0 = a[k].x, x1 = a[k].y;
        qr[4 * k + 0] = q0;
        qr[4 * k + 1] = q1;
        float e0 = q0 - x0, e1 = q1 - x1;
        dsum += e0 * e0 + e1 * e1;
        atomicAdd(&embedSum[(d0 + 0) * NEMB + myIdx], x0);
        atomicAdd(&embedSum[(d0 + 1) * NEMB + myIdx], x1);
    }
    #pragma unroll
    for (int off = 16; off >= 1; off >>= 1)
        dsum += __shfl_xor(dsum, off, 32);
    if (lane == 0) atomicAdd(diffAcc, dsum);
}

// ---------------------------------------------------------------------------
// Kernel 2: EMA buffer updates + diff finalize. One block of 512 threads.
// ---------------------------------------------------------------------------
__global__ __launch_bounds__(512) void vq_update_kernel(
    const float* __restrict__ clusterSize,
    const float* __restrict__ embedAvg,
    const float* __restrict__ hist,
    const float* __restrict__ embedSum,
    const float* __restrict__ diffAcc,
    float* __restrict__ outDiff,
    float* __restrict__ outEmbed,
    float* __restrict__ outCS,
    float* __restrict__ outEA,
    float invCount) {

    __shared__ float red[512];
    const int j = threadIdx.x;

    float ncs = clusterSize[j] * 0.99f + 0.01f * hist[j];
    outCS[j] = ncs;
    red[j] = ncs;
    __syncthreads();
    for (int s = 256; s > 0; s >>= 1) {
        if (j < s) red[j] += red[j + s];
        __syncthreads();
    }
    const float n   = red[0];
    const float csn = (ncs + 1e-5f) / (n + (float)NEMB * 1e-5f) * n;

    #pragma unroll 4
    for (int d = 0; d < DIMV; ++d) {
        int   idx = d * NEMB + j;
        float ea  = embedAvg[idx] * 0.99f + 0.01f * embedSum[idx];
        outEA[idx]    = ea;
        outEmbed[idx] = ea / csn;
    }
    if (j == 0) outDiff[0] = diffAcc[0] * invCount;
}

// ---------------------------------------------------------------------------
extern "C" void kernel_launch(void* const* d_in, const int* in_sizes, int n_in,
                              void* d_out, int out_size, void* d_ws, size_t ws_size,
                              hipStream_t stream) {
    const float* x           = (const float*)d_in[0];
    const float* embed       = (const float*)d_in[1];
    const float* clusterSize = (const float*)d_in[2];
    const float* embedAvg    = (const float*)d_in[3];

    const int N = in_sizes[0] / DIMV;   // 131072 flattened rows

    // Output layout: quantize_st | diff | embed_ind | new_embed | new_cluster_size | new_embed_avg
    float* out      = (float*)d_out;
    float* outQ     = out;
    float* outDiff  = outQ + (size_t)N * DIMV;
    float* outInd   = outDiff + 1;
    float* outEmbed = outInd + N;
    float* outCS    = outEmbed + DIMV * NEMB;
    float* outEA    = outCS + NEMB;

    // Workspace: eSq[512] | hist[512] | embedSum[64*512] | diffAcc[1]
    float* ws       = (float*)d_ws;
    float* eSq      = ws;
    float* hist     = ws + NEMB;
    float* embedSum = ws + 2 * NEMB;
    float* diffAcc  = ws + 2 * NEMB + DIMV * NEMB;

    // zero the accumulated portions each call (capture-safe memset node)
    hipMemsetAsync(hist, 0, (NEMB + DIMV * NEMB + 1) * sizeof(float), stream);

    vq_esq_kernel<<<1, 512, 0, stream>>>(embed, eSq);
    vq_dist_kernel<<<N / (WAVES * 16), 256, 0, stream>>>(
        x, embed, eSq, hist, embedSum, diffAcc, outQ, outInd);
    vq_update_kernel<<<1, 512, 0, stream>>>(
        clusterSize, embedAvg, hist, embedSum, diffAcc,
        outDiff, outEmbed, outCS, outEA, 1.0f / (float)((size_t)N * DIMV));
}